// SP_MoE_loss_60816736911483
// MI455X (gfx1250) — compile-verified
//
#include <hip/hip_runtime.h>

typedef float v2f __attribute__((ext_vector_type(2)));
typedef float v8f __attribute__((ext_vector_type(8)));

#define LAYERS 12
#define NTOK   4096
#define DEXP   16
#define BLK_PER_LAYER 16
#define TOK_PER_BLOCK 256
#define WAVES_PER_BLOCK 8

// Raw hardware transcendentals (safe here: exp args <= 0, log args >= 1e-9).
__device__ __forceinline__ float hw_exp(float x) {
    return __builtin_amdgcn_exp2f(x * 1.44269504088896340736f);   // v_exp_f32
}
__device__ __forceinline__ float hw_log(float x) {
    return __builtin_amdgcn_logf(x) * 0.69314718055994530942f;    // v_log_f32
}
__device__ __forceinline__ float hw_rcp(float x) {
    return __builtin_amdgcn_rcpf(x);                              // v_rcp_f32
}

// One token per thread: row-per-lane softmax (no cross-lane ops in hot path),
// then a per-wave WMMA column reduction of the 32x16 tile through LDS.
__global__ __launch_bounds__(256) void moe_partial_kernel(const float* __restrict__ noises,
                                                          float* __restrict__ ws_sp,
                                                          float* __restrict__ ws_lq,
                                                          float* __restrict__ ws_dg) {
    const int l    = blockIdx.x >> 4;        // layer
    const int blk  = blockIdx.x & 15;        // block within layer
    const int tid  = threadIdx.x;
    const int w    = tid >> 5;               // wave id (wave32)
    const int lane = tid & 31;
    const int d    = lane & 15;              // expert column (tail phase)
    const int h    = lane >> 4;              // half-wave (tail phase)

    const int t = blk * TOK_PER_BLOCK + tid; // this thread's token
    const float* __restrict__ rowp = noises + ((size_t)l * NTOK + t) * DEXP;

    // ---- hot path: 4x b128 loads + in-lane softmax/log, zero DS ops ----
    float x[DEXP];
    #pragma unroll
    for (int q = 0; q < 4; ++q) {
        float4 v = ((const float4*)rowp)[q];
        x[q * 4 + 0] = v.x; x[q * 4 + 1] = v.y;
        x[q * 4 + 2] = v.z; x[q * 4 + 3] = v.w;
    }

    float m = x[0];
    #pragma unroll
    for (int i = 1; i < DEXP; ++i) m = fmaxf(m, x[i]);

    float e[DEXP];
    float s = 0.f;
    #pragma unroll
    for (int i = 0; i < DEXP; ++i) { e[i] = hw_exp(x[i] - m); s += e[i]; }
    const float rs = hw_rcp(s);

    float p[DEXP], lq[DEXP], dg[DEXP];
    #pragma unroll
    for (int i = 0; i < DEXP; ++i) {
        p[i]  = e[i] * rs;
        lq[i] = hw_log(p[i] + 1e-9f);
        dg[i] = p[i] * lq[i];
    }

    // ---- exchange rows -> columns through LDS (float4 stores) ----
    __shared__ float lds_p[TOK_PER_BLOCK][DEXP];
    __shared__ float lds_l[TOK_PER_BLOCK][DEXP];
    __shared__ float lds_g[TOK_PER_BLOCK][DEXP];
    #pragma unroll
    for (int q = 0; q < 4; ++q) {
        *(float4*)&lds_p[tid][q * 4] = make_float4(p[q*4+0],  p[q*4+1],  p[q*4+2],  p[q*4+3]);
        *(float4*)&lds_l[tid][q * 4] = make_float4(lq[q*4+0], lq[q*4+1], lq[q*4+2], lq[q*4+3]);
        *(float4*)&lds_g[tid][q * 4] = make_float4(dg[q*4+0], dg[q*4+1], dg[q*4+2], dg[q*4+3]);
    }
    __syncthreads();

    // ---- per-wave WMMA column reduction of its own 32-token tile ----
    v8f acc_sp = {}, acc_lq = {}, acc_dg = {};
    const v2f ones = {1.0f, 1.0f};
    const int wtok = w * 32;
    #pragma unroll
    for (int j = 0; j < 8; ++j) {            // 4 tokens per round
        const int r0 = wtok + j * 4 + h * 2; // halves cover {t0,t1} / {t2,t3}
        v2f bp, bl, bd;
        bp[0] = lds_p[r0][d];  bp[1] = lds_p[r0 + 1][d];
        bl[0] = lds_l[r0][d];  bl[1] = lds_l[r0 + 1][d];
        bd[0] = lds_g[r0][d];  bd[1] = lds_g[r0 + 1][d];
        // D = ones(16x4) x B(4x16) + C : every row of D = column sums of B
        acc_sp = __builtin_amdgcn_wmma_f32_16x16x4_f32(false, ones, false, bp, (short)0, acc_sp, false, false);
        acc_lq = __builtin_amdgcn_wmma_f32_16x16x4_f32(false, ones, false, bl, (short)0, acc_lq, false, false);
        acc_dg = __builtin_amdgcn_wmma_f32_16x16x4_f32(false, ones, false, bd, (short)0, acc_dg, false, false);
    }

    // ---- fold 8 waves, store deterministic block partials ----
    __shared__ float red[3][WAVES_PER_BLOCK][DEXP];
    if (lane < 16) {                         // lanes 0-15: (M=0, N=lane) of D
        red[0][w][d] = acc_sp[0];
        red[1][w][d] = acc_lq[0];
        red[2][w][d] = acc_dg[0];
    }
    __syncthreads();

    if (tid < 16) {
        float sp = 0.f, sl = 0.f, sg = 0.f;
        #pragma unroll
        for (int i = 0; i < WAVES_PER_BLOCK; ++i) {
            sp += red[0][i][tid];
            sl += red[1][i][tid];
            sg += red[2][i][tid];
        }
        const int o = (l * BLK_PER_LAYER + blk) * DEXP + tid;
        ws_sp[o] = sp;
        ws_lq[o] = sl;
        ws_dg[o] = sg;
    }
}

// Fold block partials, apply the factored loss formula, emit the scalar.
__global__ __launch_bounds__(256) void moe_final_kernel(const float* __restrict__ ws_sp,
                                                        const float* __restrict__ ws_lq,
                                                        const float* __restrict__ ws_dg,
                                                        float* __restrict__ out) {
    __shared__ float Sp[LAYERS][DEXP], Sl[LAYERS][DEXP], Dg[LAYERS][DEXP];
    __shared__ float layerCE[LAYERS], layerDG[LAYERS], layerT2[LAYERS];
    const int t = threadIdx.x;

    if (t < LAYERS * DEXP) {
        const int l = t >> 4, d = t & 15;
        float sp = 0.f, lq = 0.f, dg = 0.f;
        for (int b = 0; b < BLK_PER_LAYER; ++b) {
            const int o = (l * BLK_PER_LAYER + b) * DEXP + d;
            sp += ws_sp[o];
            lq += ws_lq[o];
            dg += ws_dg[o];
        }
        Sp[l][d] = sp; Sl[l][d] = lq; Dg[l][d] = dg;
    }
    __syncthreads();

    if (t < LAYERS) {
        float dot = 0.f, dsum = 0.f, denom = 0.f;
        for (int d = 0; d < DEXP; ++d) {
            dot   += Sp[t][d] * Sl[t][d];   // ce.sum() for this layer
            dsum  += Dg[t][d];              // trace for this layer
            denom += Sp[t][d];              // = N * sum(pm)
        }
        float t2 = 0.f;
        const float invN = 1.0f / (float)NTOK;
        const float dpm  = denom * invN;    // pm normalizer
        for (int d = 0; d < DEXP; ++d) {
            float pmn = (Sp[t][d] * invN) / dpm;
            t2 += hw_log(pmn + 1e-8f);
        }
        layerCE[t] = dot;
        layerDG[t] = dsum;
        layerT2[t] = -t2 / (float)DEXP;
    }
    __syncthreads();

    if (t == 0) {
        float tot = 0.f, dg = 0.f, t2 = 0.f;
        for (int l = 0; l < LAYERS; ++l) {
            tot += layerCE[l];
            dg  += layerDG[l];
            t2  += layerT2[l];
        }
        const float aux1 = (tot - dg) / (2.0f * (float)LAYERS);
        const float aux2 = t2 / (float)LAYERS;
        out[0] = 0.01f * (1.0f * aux1 + aux2);   // ALPHA * (BETA*aux1 + aux2)
    }
}

extern "C" void kernel_launch(void* const* d_in, const int* in_sizes, int n_in,
                              void* d_out, int out_size, void* d_ws, size_t ws_size,
                              hipStream_t stream) {
    const float* noises = (const float*)d_in[0];
    float* ws    = (float*)d_ws;
    float* ws_sp = ws;
    float* ws_lq = ws + LAYERS * BLK_PER_LAYER * DEXP;
    float* ws_dg = ws + 2 * LAYERS * BLK_PER_LAYER * DEXP;

    moe_partial_kernel<<<dim3(LAYERS * BLK_PER_LAYER), dim3(256), 0, stream>>>(
        noises, ws_sp, ws_lq, ws_dg);
    moe_final_kernel<<<dim3(1), dim3(256), 0, stream>>>(
        ws_sp, ws_lq, ws_dg, (float*)d_out);
}